// GradientProcessor_76484777607756
// MI455X (gfx1250) — compile-verified
//
#include <hip/hip_runtime.h>
#include <hip/hip_bf16.h>

#define IMG   640
#define CH    3
#define NB    16
#define NP    32
#define NPATCH (NB * NP)          // 512
#define OUT_ELEMS (64 * 64 * 3)   // 12288
#define NPOS  (64 * 64)           // 4096 (y,x) positions
#define T1    256                 // threads per block, stage 1
#define PPT   16                  // positions per thread (256*16 = 4096)

typedef __attribute__((ext_vector_type(2))) float v2f;
typedef __attribute__((ext_vector_type(8))) float v8f;

// ---------------------------------------------------------------------------
// Stage 1: each block accumulates `ppg` patches (fixed serial order ->
// deterministic) into registers, then writes one partial 64x64x3 image to ws.
// Per-patch interpolation tables (offsets + weights) are built once in LDS;
// each thread then handles 16 (y,x) positions x 3 channels, so every tap is
// a load of 3 consecutive floats (vectorizable) and all floor/clip math is
// out of the inner loop.
// ---------------------------------------------------------------------------
__global__ __launch_bounds__(T1)
void stage1_patches(const float* __restrict__ grad,
                    const int*   __restrict__ boxes,
                    float*       __restrict__ ws,
                    int ppg)
{
    __shared__ int   sx0[64], sx1[64], sy0[64], sy1[64];
    __shared__ float swx[64], swy[64];

    const int t = threadIdx.x;
    const int g = blockIdx.x;

    float acc[PPT][CH];
#pragma unroll
    for (int j = 0; j < PPT; ++j)
#pragma unroll
        for (int c = 0; c < CH; ++c) acc[j][c] = 0.0f;

    const int pid0 = g * ppg;
    for (int pp = 0; pp < ppg; ++pp) {
        const int pid  = pid0 + pp;
        const int b    = pid >> 5;              // / NP
        const int ymin = boxes[4 * pid + 0];
        const int xmin = boxes[4 * pid + 1];
        const int bh   = boxes[4 * pid + 2];
        const int bw   = boxes[4 * pid + 3];
        const float scale_y = (float)bh * (1.0f / 64.0f);
        const float scale_x = (float)bw * (1.0f / 64.0f);
        const float* imgb = grad + (size_t)b * IMG * IMG * CH;

        // Build interpolation tables (TF2 bilinear, half-pixel centers).
        if (t < 64) {
            const float s  = ((float)t + 0.5f) * scale_x - 0.5f;
            const float f  = floorf(s);
            const int   i0 = (int)f;
            const int   x0 = min(max(i0,     0), bw - 1);
            const int   x1 = min(max(i0 + 1, 0), bw - 1);
            sx0[t] = (xmin + x0) * CH;
            sx1[t] = (xmin + x1) * CH;
            swx[t] = s - f;
        } else if (t < 128) {
            const int   ty = t - 64;
            const float s  = ((float)ty + 0.5f) * scale_y - 0.5f;
            const float f  = floorf(s);
            const int   i0 = (int)f;
            const int   y0 = min(max(i0,     0), bh - 1);
            const int   y1 = min(max(i0 + 1, 0), bh - 1);
            sy0[ty] = (ymin + y0) * IMG * CH;
            sy1[ty] = (ymin + y1) * IMG * CH;
            swy[ty] = s - f;
        }
        __syncthreads();

#pragma unroll
        for (int j = 0; j < PPT; ++j) {
            const int pos = t + j * T1;         // (y,x) position id
            const int x   = pos & 63;
            const int y   = pos >> 6;

            const int   ox0  = sx0[x];
            const int   ox1  = sx1[x];
            const float wx   = swx[x];
            const float omwx = 1.0f - wx;
            const int   oy0  = sy0[y];
            const int   oy1  = sy1[y];
            const float wy   = swy[y];
            const float omwy = 1.0f - wy;

            const float* p00 = imgb + oy0 + ox0;
            const float* p01 = imgb + oy0 + ox1;
            const float* p10 = imgb + oy1 + ox0;
            const float* p11 = imgb + oy1 + ox1;

#pragma unroll
            for (int c = 0; c < CH; ++c) {
                const float top = p00[c] * omwx + p01[c] * wx;
                const float bot = p10[c] * omwx + p11[c] * wx;
                acc[j][c] += top * omwy + bot * wy;
            }
        }
        __syncthreads();   // tables rebuilt next patch
    }

    float* wsg = ws + (size_t)g * OUT_ELEMS;
#pragma unroll
    for (int j = 0; j < PPT; ++j) {
        const int pos = t + j * T1;
#pragma unroll
        for (int c = 0; c < CH; ++c) wsg[pos * CH + c] = acc[j][c];
    }
}

// ---------------------------------------------------------------------------
// Stage 2: deterministic reduction over `npart` partial images using
// V_WMMA_F32_16X16X4_F32.  A-matrix holds data (A[m][k] = ws[k0+k][m],
// documented 16x4 f32 A layout), B = all-ones (layout invariant), so
// D[m][*] = sum_k A[m][k].  Each wave reduces 16 output elements over all
// partials via npart/4 chained WMMAs; rows of D extracted per the documented
// 16x16 f32 C/D layout.
// ---------------------------------------------------------------------------
__global__ __launch_bounds__(256)
void stage2_wmma_reduce(const float* __restrict__ ws,
                        float*       __restrict__ out,
                        int npart)
{
    const int lane   = threadIdx.x & 31;
    const int waveId = blockIdx.x * 8 + (threadIdx.x >> 5);
    const int m0     = waveId * 16;
    const int mA     = m0 + (lane & 15);
    const int khalf  = (lane >> 4) * 2;     // lanes 0-15: K=0,1  lanes 16-31: K=2,3

    v8f c = {};
    v2f ones = {1.0f, 1.0f};

    for (int k0 = 0; k0 < npart; k0 += 4) {
        v2f a;
        a.x = ws[(size_t)(k0 + khalf)     * OUT_ELEMS + mA];
        a.y = ws[(size_t)(k0 + khalf + 1) * OUT_ELEMS + mA];
        // (neg_a, A, neg_b, B, c_mod, C, reuse_a, reuse_b)
        c = __builtin_amdgcn_wmma_f32_16x16x4_f32(
                false, a, false, ones, (short)0, c, false, false);
    }

    // D row r lives in VGPR r: lanes 0-15 -> M=r, lanes 16-31 -> M=8+r.
    if ((lane & 15) == 0) {
        const int base = m0 + (lane >> 4) * 8;
#pragma unroll
        for (int r = 0; r < 8; ++r) out[base + r] = c[r];
    }
}

// ---------------------------------------------------------------------------
extern "C" void kernel_launch(void* const* d_in, const int* in_sizes, int n_in,
                              void* d_out, int out_size, void* d_ws, size_t ws_size,
                              hipStream_t stream)
{
    const float* grad  = (const float*)d_in[0];  // [16,640,640,3] f32
    const int*   boxes = (const int*)  d_in[1];  // [16,32,4] i32
    // d_in[2] (transform_decisions) is unused by the reference.
    float* out = (float*)d_out;                  // [64,64,3] f32
    float* wsf = (float*)d_ws;

    // Partial count adaptive to scratch size (power of two, 4..128; divides
    // 512 and is a multiple of the WMMA K-chunk of 4).
    int npart = 128;
    while (npart > 4 && (size_t)npart * OUT_ELEMS * sizeof(float) > ws_size)
        npart >>= 1;
    const int ppg = NPATCH / npart;

    hipLaunchKernelGGL(stage1_patches, dim3(npart), dim3(T1), 0, stream,
                       grad, boxes, wsf, ppg);
    hipLaunchKernelGGL(stage2_wmma_reduce, dim3(OUT_ELEMS / 16 / 8), dim3(256), 0, stream,
                       wsf, out, npart);
}